// Jump_90872918049136
// MI455X (gfx1250) — compile-verified
//
#include <hip/hip_runtime.h>
#include <math.h>

#define Bn 32
#define Tn 24
#define Nn 1024
#define Cn 3
#define Dn 64
#define Mtot (Bn*Tn*Nn)     /* 786432 tokens */
#define EPSF 1e-5f
#define LXS 68              /* padded LDS row stride: 16B-aligned + conflict-free */

typedef __attribute__((ext_vector_type(2))) float v2f;
typedef __attribute__((ext_vector_type(8))) float v8f;
typedef __attribute__((ext_vector_type(4))) int   v4i;

// ---- optional gfx1250 async global->LDS copy (guarded; falls back to sync) ----
#if defined(__has_builtin)
#  if __has_builtin(__builtin_amdgcn_global_load_async_to_lds_b128) && \
      __has_builtin(__builtin_amdgcn_s_wait_asynccnt)
#    define USE_ASYNC_COPY 1
#  endif
#endif
#ifndef USE_ASYNC_COPY
#  define USE_ASYNC_COPY 0
#endif

__device__ __forceinline__ void copy16_issue(const float* g, float* l) {
#if USE_ASYNC_COPY
  __builtin_amdgcn_global_load_async_to_lds_b128(
      (__attribute__((address_space(1))) v4i*)(g),
      (__attribute__((address_space(3))) v4i*)(l), 0, 0);
#else
  *(float4*)l = *(const float4*)g;
#endif
}
__device__ __forceinline__ void copy16_wait() {
#if USE_ASYNC_COPY
  __builtin_amdgcn_s_wait_asynccnt(0);
#endif
}

// ---------------------------------------------------------------------------
// One-shot weight swizzle: W[k][j] -> dst[((k>>1)*64 + j)*2 + (k&1)]
// makes each WMMA B-fragment (k,k+1) pair contiguous (single ds_load_b64).
// ---------------------------------------------------------------------------
__global__ __launch_bounds__(256) void k_swz(const float* __restrict__ W,
                                             float* __restrict__ dst) {
  int i = blockIdx.x * 256 + threadIdx.x;       // 4096 elements
  int k = i >> 6, j = i & 63;
  dst[(((k >> 1) * 64 + j) << 1) + (k & 1)] = W[i];
}

// ---------------------------------------------------------------------------
// Layer 0: 3->64 FC (K too small for WMMA). Gathers x[b,t,n,c] = in[b,c,n,t].
// ---------------------------------------------------------------------------
__global__ __launch_bounds__(256) void k_fc1_0(const float* __restrict__ in,
    const float* __restrict__ W, const float* __restrict__ bias,
    float* __restrict__ Y, float* __restrict__ stats) {
  __shared__ float lx[64 * 3];
  __shared__ float lsum[64], lsq[64];
  const int tid = threadIdx.x;
  const int base = blockIdx.x * 64;
  if (tid < 64) { lsum[tid] = 0.f; lsq[tid] = 0.f; }
  if (tid < 192) {
    int r = tid / 3, c = tid % 3;
    int m = base + r;
    int n = m % Nn; int bt = m / Nn; int t = bt % Tn; int b = bt / Tn;
    lx[r * 3 + c] = in[(((size_t)b * Cn + c) * Nn + n) * Tn + t];
  }
  __syncthreads();
  const int j  = tid & 63;
  const int rg = tid >> 6;
  const float w0 = W[j], w1 = W[64 + j], w2 = W[128 + j], bb = bias[j];
  float s1 = 0.f, s2 = 0.f;
  for (int r = rg * 16; r < rg * 16 + 16; ++r) {
    float y = bb + lx[r * 3] * w0 + lx[r * 3 + 1] * w1 + lx[r * 3 + 2] * w2;
    Y[(size_t)(base + r) * Dn + j] = y;
    s1 += y; s2 += y * y;
  }
  atomicAdd(&lsum[j], s1); atomicAdd(&lsq[j], s2);
  __syncthreads();
  if (tid < 64) { atomicAdd(&stats[tid], lsum[tid]); atomicAdd(&stats[64 + tid], lsq[tid]); }
}

// ---------------------------------------------------------------------------
// BN stats -> (scale, shift). scale@stats[128+], shift@stats[192+]
// ---------------------------------------------------------------------------
__global__ void k_finalize(float* __restrict__ stats, const float* __restrict__ g,
                           const float* __restrict__ be, int nch) {
  int j = threadIdx.x;
  if (j < nch) {
    const float invM = 1.f / (float)Mtot;
    float mu  = stats[j] * invM;
    float var = stats[64 + j] * invM - mu * mu;
    float sc  = g[j] * rsqrtf(var + EPSF);
    stats[128 + j] = sc;
    stats[192 + j] = be[j] - mu * sc;
  }
}

// ---------------------------------------------------------------------------
// WMMA core pieces (fp32 16x16x4), shared by single and fused-qkv GEMMs.
// ---------------------------------------------------------------------------
struct Acc4 { v8f a0, a1, a2, a3; };

__device__ __forceinline__ Acc4 gemm_core(const float* __restrict__ lx,
                                          const float* __restrict__ lws,
                                          int wave, int lane) {
  const int r0 = wave * 16;
  const int lmod = lane & 15, lhi = lane >> 4;
  const int koff = lhi * 2;                       // A: lanes16-31 hold K=2,3
  const float* arow = lx + (r0 + lmod) * LXS;
  v8f z = {0.f,0.f,0.f,0.f,0.f,0.f,0.f,0.f};
  Acc4 A; A.a0 = z; A.a1 = z; A.a2 = z; A.a3 = z;
#pragma unroll
  for (int kb = 0; kb < 64; kb += 4) {
    v2f a = *(const v2f*)(arow + kb + koff);      // aligned b64
    const float* bb = lws + (kb / 2 + lhi) * 128 + lmod * 2;
    v2f b0 = *(const v2f*)(bb);
    v2f b1 = *(const v2f*)(bb + 32);
    v2f b2 = *(const v2f*)(bb + 64);
    v2f b3 = *(const v2f*)(bb + 96);
    A.a0 = __builtin_amdgcn_wmma_f32_16x16x4_f32(false, a, false, b0, (short)0, A.a0, false, false);
    A.a1 = __builtin_amdgcn_wmma_f32_16x16x4_f32(false, a, false, b1, (short)0, A.a1, false, false);
    A.a2 = __builtin_amdgcn_wmma_f32_16x16x4_f32(false, a, false, b2, (short)0, A.a2, false, false);
    A.a3 = __builtin_amdgcn_wmma_f32_16x16x4_f32(false, a, false, b3, (short)0, A.a3, false, false);
  }
  return A;
}

__device__ __forceinline__ void gemm_epilogue(const Acc4& A,
    const float* __restrict__ lb, float* __restrict__ obuf,
    float* __restrict__ lsum, float* __restrict__ lsq, int wave, int lane) {
  const int r0 = wave * 16;
  const int lmod = lane & 15, lhi = lane >> 4;
  const v8f accs[4] = {A.a0, A.a1, A.a2, A.a3};
#pragma unroll
  for (int nt = 0; nt < 4; ++nt) {
    const int col = nt * 16 + lmod;
    const float bn = lb[col];
    float p1 = 0.f, p2 = 0.f;
#pragma unroll
    for (int jj = 0; jj < 8; ++jj) {              // C/D: VGPR jj = rows {jj | jj+8}
      int row = r0 + jj + lhi * 8;
      float y = accs[nt][jj] + bn;
      obuf[row * LXS + col] = y;
      p1 += y; p2 += y * y;
    }
    atomicAdd(&lsum[col], p1); atomicAdd(&lsq[col], p2);
  }
}

__device__ __forceinline__ void stage_x(const float* __restrict__ X, size_t rowbase,
    float* __restrict__ lx, const float* lsc, const float* lsh,
    int haspre, int prerelu, int tid) {
  for (int i4 = tid; i4 < 2048; i4 += 256) {
    int r = i4 >> 4, c4 = (i4 & 15) << 2;
    float4 xv = *(const float4*)(X + rowbase * 64 + (size_t)i4 * 4);
    if (haspre) {
      xv.x = xv.x * lsc[c4 + 0] + lsh[c4 + 0];
      xv.y = xv.y * lsc[c4 + 1] + lsh[c4 + 1];
      xv.z = xv.z * lsc[c4 + 2] + lsh[c4 + 2];
      xv.w = xv.w * lsc[c4 + 3] + lsh[c4 + 3];
      if (prerelu) {
        xv.x = fmaxf(xv.x, 0.f); xv.y = fmaxf(xv.y, 0.f);
        xv.z = fmaxf(xv.z, 0.f); xv.w = fmaxf(xv.w, 0.f);
      }
    }
    *(float4*)(lx + r * LXS + c4) = xv;
  }
}

__device__ __forceinline__ void store_tile128(const float* __restrict__ obuf,
    float* __restrict__ Y, size_t rowbase, int tid) {
  for (int i4 = tid; i4 < 2048; i4 += 256) {
    int r = i4 >> 4, c4 = (i4 & 15) << 2;
    *(float4*)(Y + rowbase * 64 + (size_t)i4 * 4) = *(const float4*)(obuf + r * LXS + c4);
  }
}

// ---------------------------------------------------------------------------
// Single-output WMMA GEMM: Y[M,64] = BN(X) @ W + bias, fused output stats.
// ---------------------------------------------------------------------------
__global__ __launch_bounds__(256) void k_gemm64(const float* __restrict__ X,
    const float* __restrict__ Wsw, const float* __restrict__ bias,
    const float* __restrict__ prestats, int prerelu,
    float* __restrict__ Y, float* __restrict__ stats) {
  __shared__ float lx[128 * LXS];
  __shared__ float lws[64 * 64];
  __shared__ float lb[64], lsc[64], lsh[64];
  __shared__ float lsum[64], lsq[64];
  const int tid = threadIdx.x;
  const size_t rowbase = (size_t)blockIdx.x * 128;
  const int haspre = (prestats != nullptr);

  for (int i4 = tid; i4 < 1024; i4 += 256) copy16_issue(Wsw + (size_t)i4 * 4, lws + i4 * 4);
  if (tid < 64) {
    lb[tid] = bias[tid];
    lsum[tid] = 0.f; lsq[tid] = 0.f;
    if (haspre) { lsc[tid] = prestats[128 + tid]; lsh[tid] = prestats[192 + tid]; }
  }
  __syncthreads();
  stage_x(X, rowbase, lx, lsc, lsh, haspre, prerelu, tid);
  copy16_wait();
  __syncthreads();

  const int wave = tid >> 5, lane = tid & 31;
  Acc4 A = gemm_core(lx, lws, wave, lane);
  __syncthreads();                                // all lx reads complete
  gemm_epilogue(A, lb, lx /* reuse as obuf */, lsum, lsq, wave, lane);
  __syncthreads();
  store_tile128(lx, Y, rowbase, tid);
  if (tid < 64) { atomicAdd(&stats[tid], lsum[tid]); atomicAdd(&stats[64 + tid], lsq[tid]); }
}

// ---------------------------------------------------------------------------
// Fused q/k/v WMMA GEMM: stage y1 once, run 3 weight sets.
// ---------------------------------------------------------------------------
__global__ __launch_bounds__(256) void k_gemm64_qkv(const float* __restrict__ X,
    const float* __restrict__ Wq, const float* __restrict__ Wk, const float* __restrict__ Wv,
    const float* __restrict__ bq, const float* __restrict__ bk, const float* __restrict__ bv,
    const float* __restrict__ prestats,
    float* __restrict__ Yq, float* __restrict__ Yk, float* __restrict__ Yv,
    float* __restrict__ Sq, float* __restrict__ Sk, float* __restrict__ Sv) {
  __shared__ float lx[128 * LXS];
  __shared__ float obuf[128 * LXS];
  __shared__ float lws[3 * 64 * 64];
  __shared__ float lb[192], lsc[64], lsh[64];
  __shared__ float lsum[192], lsq[192];
  const int tid = threadIdx.x;
  const size_t rowbase = (size_t)blockIdx.x * 128;

  for (int i4 = tid; i4 < 1024; i4 += 256) {
    copy16_issue(Wq + (size_t)i4 * 4, lws + i4 * 4);
    copy16_issue(Wk + (size_t)i4 * 4, lws + 4096 + i4 * 4);
    copy16_issue(Wv + (size_t)i4 * 4, lws + 8192 + i4 * 4);
  }
  if (tid < 64) {
    lb[tid] = bq[tid]; lb[64 + tid] = bk[tid]; lb[128 + tid] = bv[tid];
    lsc[tid] = prestats[128 + tid]; lsh[tid] = prestats[192 + tid];
  }
  if (tid < 192) { lsum[tid] = 0.f; lsq[tid] = 0.f; }
  __syncthreads();
  stage_x(X, rowbase, lx, lsc, lsh, 1, 0 /* fc1_1 has no relu */, tid);
  copy16_wait();
  __syncthreads();

  const int wave = tid >> 5, lane = tid & 31;
  float* Yp[3] = {Yq, Yk, Yv};
#pragma unroll
  for (int s = 0; s < 3; ++s) {
    Acc4 A = gemm_core(lx, lws + s * 4096, wave, lane);
    gemm_epilogue(A, lb + s * 64, obuf, lsum + s * 64, lsq + s * 64, wave, lane);
    __syncthreads();
    store_tile128(obuf, Yp[s], rowbase, tid);
    __syncthreads();                              // obuf reused next set
  }
  if (tid < 192) {
    float* Sp = (tid < 64) ? Sq : (tid < 128) ? Sk : Sv;
    int j = tid & 63;
    atomicAdd(&Sp[j], lsum[tid]); atomicAdd(&Sp[64 + j], lsq[tid]);
  }
}

// ---------------------------------------------------------------------------
// Attention: one block per (b,n); 8 waves = 8 heads; lane = query row (T=24).
// ---------------------------------------------------------------------------
__global__ __launch_bounds__(256) void k_attn(const float* __restrict__ Q,
    const float* __restrict__ Kb, const float* __restrict__ V,
    const float* __restrict__ sq, const float* __restrict__ sk,
    const float* __restrict__ sv, float* __restrict__ O) {
  __shared__ float qs[24 * 65], ks[24 * 65], vs[24 * 65];
  const int tid = threadIdx.x;
  const int b = blockIdx.x >> 10;
  const int n = blockIdx.x & 1023;

  for (int i4 = tid; i4 < 24 * 16; i4 += 256) {
    int r = i4 >> 4, c4 = (i4 & 15) << 2;
    size_t m = ((size_t)(b * Tn + r) * Nn + n);
    float4 q4 = *(const float4*)(Q  + m * 64 + c4);
    float4 k4 = *(const float4*)(Kb + m * 64 + c4);
    float4 v4 = *(const float4*)(V  + m * 64 + c4);
    qs[r * 65 + c4 + 0] = fmaxf(q4.x * sq[128 + c4 + 0] + sq[192 + c4 + 0], 0.f);
    qs[r * 65 + c4 + 1] = fmaxf(q4.y * sq[128 + c4 + 1] + sq[192 + c4 + 1], 0.f);
    qs[r * 65 + c4 + 2] = fmaxf(q4.z * sq[128 + c4 + 2] + sq[192 + c4 + 2], 0.f);
    qs[r * 65 + c4 + 3] = fmaxf(q4.w * sq[128 + c4 + 3] + sq[192 + c4 + 3], 0.f);
    ks[r * 65 + c4 + 0] = fmaxf(k4.x * sk[128 + c4 + 0] + sk[192 + c4 + 0], 0.f);
    ks[r * 65 + c4 + 1] = fmaxf(k4.y * sk[128 + c4 + 1] + sk[192 + c4 + 1], 0.f);
    ks[r * 65 + c4 + 2] = fmaxf(k4.z * sk[128 + c4 + 2] + sk[192 + c4 + 2], 0.f);
    ks[r * 65 + c4 + 3] = fmaxf(k4.w * sk[128 + c4 + 3] + sk[192 + c4 + 3], 0.f);
    vs[r * 65 + c4 + 0] = fmaxf(v4.x * sv[128 + c4 + 0] + sv[192 + c4 + 0], 0.f);
    vs[r * 65 + c4 + 1] = fmaxf(v4.y * sv[128 + c4 + 1] + sv[192 + c4 + 1], 0.f);
    vs[r * 65 + c4 + 2] = fmaxf(v4.z * sv[128 + c4 + 2] + sv[192 + c4 + 2], 0.f);
    vs[r * 65 + c4 + 3] = fmaxf(v4.w * sv[128 + c4 + 3] + sv[192 + c4 + 3], 0.f);
  }
  __syncthreads();

  const int wave = tid >> 5, lane = tid & 31;
  if (lane < 24) {
    const int hc = wave * 8;
    float qr[8];
#pragma unroll
    for (int d0 = 0; d0 < 8; ++d0) qr[d0] = qs[lane * 65 + hc + d0];
    float sc[24]; float mx = -1e30f;
#pragma unroll
    for (int kk = 0; kk < 24; ++kk) {
      float s = 0.f;
#pragma unroll
      for (int d0 = 0; d0 < 8; ++d0) s += qr[d0] * ks[kk * 65 + hc + d0];
      s *= 0.35355339059327373f;
      sc[kk] = s; mx = fmaxf(mx, s);
    }
    float ssum = 0.f;
#pragma unroll
    for (int kk = 0; kk < 24; ++kk) { float e = __expf(sc[kk] - mx); sc[kk] = e; ssum += e; }
    const float inv = 1.f / ssum;
    float o[8] = {0.f,0.f,0.f,0.f,0.f,0.f,0.f,0.f};
#pragma unroll
    for (int kk = 0; kk < 24; ++kk) {
      float a = sc[kk] * inv;
#pragma unroll
      for (int d0 = 0; d0 < 8; ++d0) o[d0] += a * vs[kk * 65 + hc + d0];
    }
    size_t m = ((size_t)(b * Tn + lane) * Nn + n);
    *(float4*)(O + m * 64 + hc)     = make_float4(o[0], o[1], o[2], o[3]);
    *(float4*)(O + m * 64 + hc + 4) = make_float4(o[4], o[5], o[6], o[7]);
  }
}

// ---------------------------------------------------------------------------
// fc2_1: 64 -> 1 (only channel 0 survives).
// ---------------------------------------------------------------------------
__global__ __launch_bounds__(256) void k_fc2_1(const float* __restrict__ X,
    const float* __restrict__ W, const float* __restrict__ bias,
    const float* __restrict__ pre, float* __restrict__ y5, float* __restrict__ stats) {
  __shared__ float ld[128 * 65];
  __shared__ float wcol[64];
  __shared__ float bsum[2];
  const int tid = threadIdx.x;
  const size_t rowbase = (size_t)blockIdx.x * 128;
  if (tid < 64) wcol[tid] = W[tid * 2];
  if (tid < 2)  bsum[tid] = 0.f;
  for (int i4 = tid; i4 < 2048; i4 += 256) {
    int r = i4 >> 4, c4 = (i4 & 15) << 2;
    float4 xv = *(const float4*)(X + rowbase * 64 + (size_t)i4 * 4);
    ld[r * 65 + c4 + 0] = fmaxf(xv.x * pre[128 + c4 + 0] + pre[192 + c4 + 0], 0.f);
    ld[r * 65 + c4 + 1] = fmaxf(xv.y * pre[128 + c4 + 1] + pre[192 + c4 + 1], 0.f);
    ld[r * 65 + c4 + 2] = fmaxf(xv.z * pre[128 + c4 + 2] + pre[192 + c4 + 2], 0.f);
    ld[r * 65 + c4 + 3] = fmaxf(xv.w * pre[128 + c4 + 3] + pre[192 + c4 + 3], 0.f);
  }
  __syncthreads();
  if (tid < 128) {
    float y = bias[0];
#pragma unroll
    for (int c = 0; c < 64; ++c) y += ld[tid * 65 + c] * wcol[c];
    y5[rowbase + tid] = y;
    atomicAdd(&bsum[0], y); atomicAdd(&bsum[1], y * y);
  }
  __syncthreads();
  if (tid == 0) { atomicAdd(&stats[0], bsum[0]); atomicAdd(&stats[64], bsum[1]); }
}

// ---------------------------------------------------------------------------
// Final BN + transpose [B,T,N] -> [T,B,N]  (float4 both sides)
// ---------------------------------------------------------------------------
__global__ __launch_bounds__(256) void k_out(const float* __restrict__ y5,
    const float* __restrict__ stats, float* __restrict__ out) {
  const int i4 = blockIdx.x * 256 + threadIdx.x;  // over (T*B*N)/4
  const int idx = i4 * 4;
  const int n4 = idx & (Nn - 1);
  const int r = idx >> 10;
  const int b = r % Bn;
  const int t = r / Bn;
  const size_t m = ((size_t)(b * Tn + t) * Nn + n4);
  const float sc = stats[128], sh = stats[192];
  float4 v = *(const float4*)(y5 + m);
  *(float4*)(out + idx) = make_float4(v.x * sc + sh, v.y * sc + sh, v.z * sc + sh, v.w * sc + sh);
}

// ---------------------------------------------------------------------------
extern "C" void kernel_launch(void* const* d_in, const int* in_sizes, int n_in,
                              void* d_out, int out_size, void* d_ws, size_t ws_size,
                              hipStream_t stream) {
  const float* inputs  = (const float*)d_in[0];
  const float* fc1_w0  = (const float*)d_in[2];
  const float* fc1_b0  = (const float*)d_in[3];
  const float* fc1_g0  = (const float*)d_in[4];
  const float* fc1_be0 = (const float*)d_in[5];
  const float* fc1_w1  = (const float*)d_in[6];
  const float* fc1_b1  = (const float*)d_in[7];
  const float* fc1_g1  = (const float*)d_in[8];
  const float* fc1_be1 = (const float*)d_in[9];
  const float* qw = (const float*)d_in[10]; const float* qb = (const float*)d_in[11];
  const float* qg = (const float*)d_in[12]; const float* qbe = (const float*)d_in[13];
  const float* kw = (const float*)d_in[14]; const float* kb = (const float*)d_in[15];
  const float* kg = (const float*)d_in[16]; const float* kbe = (const float*)d_in[17];
  const float* vw = (const float*)d_in[18]; const float* vb = (const float*)d_in[19];
  const float* vg = (const float*)d_in[20]; const float* vbe = (const float*)d_in[21];
  const float* ow = (const float*)d_in[22]; const float* ob = (const float*)d_in[23];
  const float* og = (const float*)d_in[24]; const float* obe = (const float*)d_in[25];
  const float* fc2_w0  = (const float*)d_in[26];
  const float* fc2_b0  = (const float*)d_in[27];
  const float* fc2_g0  = (const float*)d_in[28];
  const float* fc2_be0 = (const float*)d_in[29];
  const float* fc2_w1  = (const float*)d_in[30];
  const float* fc2_b1  = (const float*)d_in[31];
  const float* fc2_g1  = (const float*)d_in[32];
  const float* fc2_be1 = (const float*)d_in[33];

  float* ws = (float*)d_ws;
  const size_t M64 = (size_t)Mtot * 64;
  float* A  = ws;               // y0 -> later v
  float* Bu = ws + M64;         // y1 -> later o
  float* Cu = ws + 2 * M64;     // q  -> later y3
  float* Du = ws + 3 * M64;     // k  -> later y4
  float* y5 = ws + 4 * M64;     // [M]
  float* ST = y5 + Mtot;        // 8 stat slots x 256 floats
  float* S0 = ST;        float* S1 = ST + 256;  float* Sq = ST + 512;
  float* Sk = ST + 768;  float* Sv = ST + 1024; float* S3 = ST + 1280;
  float* S4 = ST + 1536; float* S5 = ST + 1792;
  float* WZ = ST + 2048;        // 6 swizzled 64x64 weights
  float* Z1 = WZ;          float* Zq = WZ + 4096;  float* Zk = WZ + 8192;
  float* Zv = WZ + 12288;  float* Zo = WZ + 16384; float* Z2 = WZ + 20480;

  (void)hipMemsetAsync(ST, 0, 8 * 256 * sizeof(float), stream);

  k_swz<<<16, 256, 0, stream>>>(fc1_w1, Z1);
  k_swz<<<16, 256, 0, stream>>>(qw, Zq);
  k_swz<<<16, 256, 0, stream>>>(kw, Zk);
  k_swz<<<16, 256, 0, stream>>>(vw, Zv);
  k_swz<<<16, 256, 0, stream>>>(ow, Zo);
  k_swz<<<16, 256, 0, stream>>>(fc2_w0, Z2);

  k_fc1_0<<<Mtot / 64, 256, 0, stream>>>(inputs, fc1_w0, fc1_b0, A, S0);
  k_finalize<<<1, 64, 0, stream>>>(S0, fc1_g0, fc1_be0, 64);

  k_gemm64<<<Mtot / 128, 256, 0, stream>>>(A, Z1, fc1_b1, S0, 1, Bu, S1);
  k_finalize<<<1, 64, 0, stream>>>(S1, fc1_g1, fc1_be1, 64);

  k_gemm64_qkv<<<Mtot / 128, 256, 0, stream>>>(Bu, Zq, Zk, Zv, qb, kb, vb, S1,
                                               Cu, Du, A, Sq, Sk, Sv);
  k_finalize<<<1, 64, 0, stream>>>(Sq, qg, qbe, 64);
  k_finalize<<<1, 64, 0, stream>>>(Sk, kg, kbe, 64);
  k_finalize<<<1, 64, 0, stream>>>(Sv, vg, vbe, 64);

  k_attn<<<Bn * Nn, 256, 0, stream>>>(Cu, Du, A, Sq, Sk, Sv, Bu);

  k_gemm64<<<Mtot / 128, 256, 0, stream>>>(Bu, Zo, ob, nullptr, 0, Cu, S3);
  k_finalize<<<1, 64, 0, stream>>>(S3, og, obe, 64);

  k_gemm64<<<Mtot / 128, 256, 0, stream>>>(Cu, Z2, fc2_b0, S3, 1, Du, S4);
  k_finalize<<<1, 64, 0, stream>>>(S4, fc2_g0, fc2_be0, 64);

  k_fc2_1<<<Mtot / 128, 256, 0, stream>>>(Du, fc2_w1, fc2_b1, S4, y5, S5);
  k_finalize<<<1, 64, 0, stream>>>(S5, fc2_g1, fc2_be1, 1);

  k_out<<<Mtot / 1024, 256, 0, stream>>>(y5, S5, (float*)d_out);
}